// SE3ContiFormer_75892072120804
// MI455X (gfx1250) — compile-verified
//
#include <hip/hip_runtime.h>
#include <math.h>

typedef float v2f __attribute__((ext_vector_type(2)));
typedef float v8f __attribute__((ext_vector_type(8)));

#define ALPHA0 0.1889822365046136f    /* 1/sqrt(28) */
#define ALPHA1 0.1666666666666667f    /* 1/6        */
#define ALPHA2 0.1767766952966369f    /* 1/sqrt(32) */
#define ALPHA_DOT 0.0545544725589981f /* 1/sqrt(336) */
#define APOT   0.1889822365046136f    /* 1/sqrt(28) */

// ---------------- CG table generation (device, double precision) ----------------
struct cdbl { double re, im; };
__device__ inline cdbl cmul(cdbl a, cdbl b){ cdbl r; r.re=a.re*b.re-a.im*b.im; r.im=a.re*b.im+a.im*b.re; return r; }
__device__ inline cdbl cadd(cdbl a, cdbl b){ cdbl r; r.re=a.re+b.re; r.im=a.im+b.im; return r; }
__device__ inline cdbl cconj(cdbl a){ cdbl r; r.re=a.re; r.im=-a.im; return r; }
__device__ inline cdbl cscale(double s, cdbl a){ cdbl r; r.re=s*a.re; r.im=s*a.im; return r; }

__device__ double dfact(int n){ double r=1.0; for(int i=2;i<=n;++i) r*=(double)i; return r; }

__device__ double su2_cg(int j1,int j2,int j3,int m1,int m2,int m3){
  if (m3 != m1+m2) return 0.0;
  int dj = j1-j2; if (dj<0) dj=-dj;
  if (j3 < dj || j3 > j1+j2) return 0.0;
  double pref = sqrt((double)(2*j3+1)*dfact(j3+j1-j2)*dfact(j3-j1+j2)*dfact(j1+j2-j3)/dfact(j1+j2+j3+1));
  pref *= sqrt(dfact(j3+m3)*dfact(j3-m3)*dfact(j1-m1)*dfact(j1+m1)*dfact(j2-m2)*dfact(j2+m2));
  double s = 0.0;
  for (int v=0; v<=j1+j2-j3; ++v){
    int den[6] = {v, j1+j2-j3-v, j1-m1-v, j2+m2-v, j3-j2+m1+v, j3-j1-m2+v};
    bool ok = true; double d = 1.0;
    for (int t=0;t<6;++t){ if(den[t]<0){ ok=false; break; } d *= dfact(den[t]); }
    if (!ok) continue;
    s += ((v&1)? -1.0 : 1.0)/d;
  }
  return pref*s;
}

__device__ void r2c_mat(int l, cdbl q[5][5]){
  int dd = 2*l+1;
  for (int i=0;i<5;++i) for(int j=0;j<5;++j){ q[i][j].re=0.0; q[i][j].im=0.0; }
  double is2 = 1.0/sqrt(2.0);
  for (int m=-l; m<0; ++m){ q[l+m][l-m].re = is2; q[l+m][l+m].im = -is2; }
  q[l][l].re = 1.0;
  for (int m=1; m<=l; ++m){
    double sg = (m&1)? -1.0 : 1.0;
    q[l+m][l+m].re = sg*is2; q[l+m][l-m].im = sg*is2;
  }
  cdbl ph; if (l==0){ph.re=1;ph.im=0;} else if(l==1){ph.re=0;ph.im=-1;} else {ph.re=-1;ph.im=0;}
  for (int i=0;i<dd;++i) for(int j=0;j<dd;++j) q[i][j] = cmul(ph, q[i][j]);
}

__device__ void real_cg_fill(int l1,int l2,int l3,float* dst){
  int d1=2*l1+1, d2=2*l2+1, d3=2*l3+1;
  cdbl q1[5][5], q2[5][5], q3[5][5];
  r2c_mat(l1,q1); r2c_mat(l2,q2); r2c_mat(l3,q3);
  double Tr[125], Ti[125]; double nr=0.0, ni=0.0;
  for (int i=0;i<d1;++i) for(int j=0;j<d2;++j) for(int kk=0;kk<d3;++kk){
    cdbl s; s.re=0; s.im=0;
    for (int a=0;a<d1;++a) for(int b=0;b<d2;++b) for(int c=0;c<d3;++c){
      double C = su2_cg(l1,l2,l3,a-l1,b-l2,c-l3);
      if (C == 0.0) continue;
      cdbl t = cmul(q1[i][a], cmul(q2[j][b], cconj(q3[kk][c])));
      s = cadd(s, cscale(C,t));
    }
    int idx = (i*d2+j)*d3+kk;
    Tr[idx]=s.re; Ti[idx]=s.im;
    nr += s.re*s.re; ni += s.im*s.im;
  }
  bool useR = (nr >= ni);
  double nrm = sqrt(useR? nr : ni);
  double inv = (nrm > 0.0)? 1.0/nrm : 0.0;
  int tot = d1*d2*d3;
  for (int idx=0; idx<tot; ++idx) dst[idx] = (float)((useR? Tr[idx] : Ti[idx])*inv);
}

__global__ void init_cg_kernel(float* cg){
  int t = threadIdx.x;
  const int L1[11]={0,1,2,0,1,1,2,0,1,2,2};
  const int L2[11]={0,1,2,1,0,2,1,2,1,0,2};
  const int L3[11]={0,0,0,1,1,1,1,2,2,2,2};
  const int OFF[11]={0,1,10,35,44,53,98,143,168,213,238};
  if (t < 11) real_cg_fill(L1[t],L2[t],L3[t], cg+OFF[t]);
}

// ---------------- misc small kernels ----------------
__global__ void fill_kernel(float* p, int n, float v){
  int i = blockIdx.x*blockDim.x + threadIdx.x;
  if (i < n) p[i] = v;
}

__global__ void feat_init_kernel(const float* __restrict__ x, const float* __restrict__ Win,
                                 float* __restrict__ feat, int Ntot, int F){
  int gid = blockIdx.x*blockDim.x + threadIdx.x;
  if (gid >= Ntot*60) return;
  int n = gid/60, c = gid%60;
  float v = 0.f;
  if (c < 16){
    float s = 0.f;
    for (int f=0; f<F; ++f) s += x[(size_t)n*F+f]*Win[f*16+c];
    v = s*(1.0f/sqrtf((float)F));
  }
  feat[gid] = v;
}

__global__ void edge_prep_kernel(const float* __restrict__ pos, const int* __restrict__ src,
                                 const int* __restrict__ dst, float* __restrict__ esh,
                                 float* __restrict__ emb, float* __restrict__ cutb, int E){
  int e = blockIdx.x*blockDim.x + threadIdx.x;
  if (e >= E) return;
  int s = src[e], d = dst[e];
  float vx = pos[(size_t)s*3+0]-pos[(size_t)d*3+0];
  float vy = pos[(size_t)s*3+1]-pos[(size_t)d*3+1];
  float vz = pos[(size_t)s*3+2]-pos[(size_t)d*3+2];
  float r = sqrtf(vx*vx+vy*vy+vz*vz);
  float inv = 1.f/fmaxf(r, 1e-9f);
  float ux=vx*inv, uy=vy*inv, uz=vz*inv;
  const float s3 = 1.7320508075688772f, s15 = 3.872983346207417f, s5 = 2.23606797749979f;
  float* sh = esh + (size_t)e*9;
  sh[0]=1.f; sh[1]=s3*ux; sh[2]=s3*uy; sh[3]=s3*uz;
  sh[4]=s15*ux*uy; sh[5]=s15*uy*uz; sh[6]=0.5f*s5*(3.f*uz*uz-1.f);
  sh[7]=s15*ux*uz; sh[8]=0.5f*s15*(ux*ux-uy*uy);
  const float step = 3.f/11.f;
  for (int i=0;i<10;++i){
    float val = 3.f*(float)(i+1)/11.f;
    float dd = (r - val)/step;
    float d2 = fminf(dd*dd, 1.f-1e-6f);
    float b = (fabsf(dd) < 1.f) ? 1.14136f*expf(2.f-2.f/(1.f-d2)) : 0.f;
    emb[(size_t)e*10+i] = b*3.1622776601683795f; // * sqrt(10)
  }
  float xcut = 10.f*(1.f - r/3.f);
  cutb[e] = (xcut > 0.f) ? expf(-1.f/fmaxf(xcut, 1e-9f)) : 0.f;
}

__global__ void q_kernel(const float* __restrict__ feat, const float* __restrict__ hq0,
                         const float* __restrict__ hq1, const float* __restrict__ hq2,
                         float* __restrict__ q, int Ntot){
  int gid = blockIdx.x*blockDim.x + threadIdx.x;
  if (gid >= Ntot*60) return;
  int n = gid/60, o = gid%60;
  const float* f = feat + (size_t)n*60;
  float s = 0.f;
  if (o < 16){
    for (int u=0;u<16;++u) s += f[u]*hq0[u*16+o];
    s *= 0.25f;
  } else if (o < 40){
    int idx=o-16, v=idx/3, i=idx%3;
    for (int u=0;u<8;++u) s += f[16+u*3+i]*hq1[u*8+v];
    s *= 0.35355339059327373f;
  } else {
    int idx=o-40, v=idx/5, i=idx%5;
    for (int u=0;u<4;++u) s += f[40+u*5+i]*hq2[u*4+v];
    s *= 0.5f;
  }
  q[gid] = s;
}

// h = silu(emb @ W1 / sqrt(10)) * scale   (scale folds fcnet's 1/sqrt(hidden))
__global__ void h_kernel(const float* __restrict__ emb, const float* __restrict__ W1,
                         float* __restrict__ h, int E, int C, float scale){
  long gid = blockIdx.x*(long)blockDim.x + threadIdx.x;
  if (gid >= (long)E*C) return;
  int e = (int)(gid/C), c = (int)(gid%C);
  float s = 0.f;
  for (int b=0;b<10;++b) s += emb[(size_t)e*10+b]*W1[b*C+c];
  s *= 0.31622776601683794f; // 1/sqrt(10)
  float sig = 1.f/(1.f+expf(-s));
  h[gid] = s*sig*scale;
}

// ---------------- fused WMMA weight-GEMM + tensor product (k / v) ----------------
// Fully compile-time-unrolled tensor-product path: no descriptor loads, no
// dynamic trip counts -> pure FMA streams against LDS.
template<int OFF1,int M1,int D1,int SHOFF,int D2,int CGOFF,int WS,int MO,int DOUT>
__device__ __forceinline__ float tp_path(const float* __restrict__ fs,
                                         const float* __restrict__ sh,
                                         const float* __restrict__ wb,
                                         const float* __restrict__ cg,
                                         int w, int kp){
  float ct[D1];
#pragma unroll
  for (int i=0;i<D1;++i){
    float s = 0.f;
#pragma unroll
    for (int j=0;j<D2;++j)
      s += cg[CGOFF + (i*D2+j)*DOUT + kp] * sh[SHOFF + j];
    ct[i] = s;
  }
  float acc = 0.f;
#pragma unroll
  for (int u=0;u<M1;++u){
    float t = 0.f;
#pragma unroll
    for (int i=0;i<D1;++i) t += fs[OFF1 + u*D1 + i]*ct[i];
    acc += wb[WS + u*MO + w]*t;
  }
  return acc;
}

__global__ __launch_bounds__(256) void tp_kv_kernel(
    const float* __restrict__ feat, const int* __restrict__ esrc,
    const float* __restrict__ esh, const float* __restrict__ h,
    const float* __restrict__ W2, const float* __restrict__ cgg,
    float* __restrict__ out, int E)
{
  __shared__ float s_wb[16*864];  // per-edge TP weights: h_tile @ W2
  __shared__ float s_fs[16*60];   // gathered source features
  __shared__ float s_sh[16*9];    // spherical harmonics
  __shared__ float s_cg[384];     // CG coefficient table (broadcast reads)
  int e0 = blockIdx.x*16;
  int tid = threadIdx.x;

  for (int i=tid; i<16*60; i+=256){
    int el=i/60, c=i%60; int ee=e0+el;
    s_fs[i] = (ee<E) ? feat[(size_t)esrc[ee]*60 + c] : 0.f;
  }
  for (int i=tid; i<16*9; i+=256){
    int el=i/9, c=i%9; int ee=e0+el;
    s_sh[i] = (ee<E) ? esh[(size_t)ee*9+c] : 0.f;
  }
  if (tid < 192){
    s_cg[tid] = cgg[tid];
    if (tid < 171) s_cg[192+tid] = cgg[192+tid];
  }

  // ---- WMMA phase: s_wb[16][864] = h[e0:e0+16, 0:32] @ W2[32, 864] ----
  int wv = tid>>5, lane = tid&31;
  int r16 = lane&15, kg = lane>>4;
  int eg = e0 + r16;
  bool vr = (eg < E);
  v2f afrag[8];
#pragma unroll
  for (int s=0;s<8;++s){
    int kb = 4*s + 2*kg;
    afrag[s].x = vr ? h[(size_t)eg*32 + kb]     : 0.f;
    afrag[s].y = vr ? h[(size_t)eg*32 + kb + 1] : 0.f;
  }
  for (int tc=wv; tc<54; tc+=8){
    v8f acc = {0.f,0.f,0.f,0.f,0.f,0.f,0.f,0.f};
#pragma unroll
    for (int s=0;s<8;++s){
      int kb = 4*s + 2*kg;
      v2f bfrag;
      bfrag.x = W2[(size_t)kb*864     + tc*16 + r16];
      bfrag.y = W2[(size_t)(kb+1)*864 + tc*16 + r16];
      acc = __builtin_amdgcn_wmma_f32_16x16x4_f32(false, afrag[s], false, bfrag,
                                                  (short)0, acc, false, false);
    }
#pragma unroll
    for (int r=0;r<8;++r)
      s_wb[(r + 8*kg)*864 + tc*16 + r16] = acc[r];
  }
  __syncthreads();

  // ---- TP phase: 16 threads per edge, each covers 4 of 60 outputs ----
  int el = tid>>4, l16 = tid&15;
  int ego = e0 + el;
  if (ego < E){
    const float* fs = s_fs + el*60;
    const float* sh = s_sh + el*9;
    const float* wb = s_wb + el*864;
    for (int rep=0; rep<4; ++rep){
      int o = l16 + 16*rep;
      if (o < 60){
        float val;
        if (o < 16){
          float a = tp_path< 0,16,1, 0,1,  0,  0,16,1>(fs, sh, wb, s_cg, o, 0)
                  + tp_path<16, 8,3, 1,3,  1,256,16,1>(fs, sh, wb, s_cg, o, 0)
                  + tp_path<40, 4,5, 4,5, 10,384,16,1>(fs, sh, wb, s_cg, o, 0);
          val = ALPHA0*a;
        } else if (o < 40){
          int idx = o-16, w = idx/3, kp = idx%3;
          float a = tp_path< 0,16,1, 1,3, 35,448, 8,3>(fs, sh, wb, s_cg, w, kp)
                  + tp_path<16, 8,3, 0,1, 44,576, 8,3>(fs, sh, wb, s_cg, w, kp)
                  + tp_path<16, 8,3, 4,5, 53,640, 8,3>(fs, sh, wb, s_cg, w, kp)
                  + tp_path<40, 4,5, 1,3, 98,704, 8,3>(fs, sh, wb, s_cg, w, kp);
          val = ALPHA1*a;
        } else {
          int idx = o-40, w = idx/5, kp = idx%5;
          float a = tp_path< 0,16,1, 4,5,143,736, 4,5>(fs, sh, wb, s_cg, w, kp)
                  + tp_path<16, 8,3, 1,3,168,800, 4,5>(fs, sh, wb, s_cg, w, kp)
                  + tp_path<40, 4,5, 0,1,213,832, 4,5>(fs, sh, wb, s_cg, w, kp)
                  + tp_path<40, 4,5, 4,5,238,848, 4,5>(fs, sh, wb, s_cg, w, kp);
          val = ALPHA2*a;
        }
        out[(size_t)ego*60 + o] = val;
      }
    }
  }
}

// ---------------- attention logits + softmax denominator ----------------
__global__ void logit_kernel(const float* __restrict__ q, const float* __restrict__ k,
                             const int* __restrict__ edst, const float* __restrict__ cutb,
                             const float* __restrict__ dw, const float* __restrict__ cg,
                             float* __restrict__ expv, float* __restrict__ z, int E){
  int e = blockIdx.x*blockDim.x + threadIdx.x;
  if (e >= E) return;
  int d = edst[e];
  const float* a = q + (size_t)d*60;
  const float* b = k + (size_t)e*60;
  float acc = 0.f;
  float c000 = cg[0];
#pragma unroll
  for (int u=0;u<16;++u){
    float au = a[u]*c000;
#pragma unroll
    for (int v=0;v<16;++v) acc += dw[u*16+v]*au*b[v];
  }
#pragma unroll
  for (int u=0;u<8;++u){
    float ar[3];
#pragma unroll
    for (int j=0;j<3;++j){
      float s = 0.f;
#pragma unroll
      for (int i=0;i<3;++i) s += a[16+u*3+i]*cg[1 + i*3+j];
      ar[j] = s;
    }
#pragma unroll
    for (int v=0;v<8;++v){
      float dm = ar[0]*b[16+v*3] + ar[1]*b[16+v*3+1] + ar[2]*b[16+v*3+2];
      acc += dw[256+u*8+v]*dm;
    }
  }
#pragma unroll
  for (int u=0;u<4;++u){
    float ar[5];
#pragma unroll
    for (int j=0;j<5;++j){
      float s = 0.f;
#pragma unroll
      for (int i=0;i<5;++i) s += a[40+u*5+i]*cg[10 + i*5+j];
      ar[j] = s;
    }
#pragma unroll
    for (int v=0;v<4;++v){
      float dm = 0.f;
#pragma unroll
      for (int j=0;j<5;++j) dm += ar[j]*b[40+v*5+j];
      acc += dw[320+u*4+v]*dm;
    }
  }
  float ev = cutb[e]*expf(acc*ALPHA_DOT);
  expv[e] = ev;
  atomicAdd(&z[d], ev);
}

__global__ void att_kernel(const float* __restrict__ v, const float* __restrict__ expv,
                           const float* __restrict__ z, const int* __restrict__ edst,
                           float* __restrict__ att, int E){
  long gid = blockIdx.x*(long)blockDim.x + threadIdx.x;
  if (gid >= (long)E*60) return;
  int e = (int)(gid/60), c = (int)(gid%60);
  int d = edst[e];
  float zd = z[d]; zd = (zd == 0.f) ? 1.f : zd;
  float al = expv[e]/zd;
  float s = sqrtf(fmaxf(al, 0.f));
  atomicAdd(&att[(size_t)d*60 + c], s*v[(size_t)e*60+c]);
}

// ---------------- potential: only channels 0..2 of lo=0 irrep are ever used ----------------
__global__ void pot_kernel(const float* __restrict__ att, const int* __restrict__ esrc,
                           const int* __restrict__ edst, const float* __restrict__ esh,
                           const float* __restrict__ hp, const float* __restrict__ W2,
                           const float* __restrict__ cutb, const float* __restrict__ cg,
                           float* __restrict__ pot3, int E){
  int gid = blockIdx.x*blockDim.x + threadIdx.x;
  if (gid >= 3*E) return;
  int e = gid/3, w = gid%3;
  const float* a  = att + (size_t)esrc[e]*60;
  const float* sh = esh + (size_t)e*9;
  float t[28];
  float c000 = cg[0];
#pragma unroll
  for (int u=0;u<16;++u) t[u] = a[u]*c000*sh[0];
#pragma unroll
  for (int u=0;u<8;++u){
    float s = 0.f;
#pragma unroll
    for (int i=0;i<3;++i)
#pragma unroll
      for (int j=0;j<3;++j)
        s += a[16+u*3+i]*cg[1 + i*3+j]*sh[1+j];
    t[16+u] = s;
  }
#pragma unroll
  for (int u=0;u<4;++u){
    float s = 0.f;
#pragma unroll
    for (int i=0;i<5;++i)
#pragma unroll
      for (int j=0;j<5;++j)
        s += a[40+u*5+i]*cg[10 + i*5+j]*sh[4+j];
    t[24+u] = s;
  }
  const float* h = hp + (size_t)e*64;
  float acc = 0.f;
  for (int c=0;c<64;++c){
    float hc = h[c];
    const float* wr = W2 + (size_t)c*1472 + w;
    float s2 = 0.f;
#pragma unroll
    for (int u=0;u<28;++u) s2 += wr[u*32]*t[u];
    acc += hc*s2;
  }
  atomicAdd(&pot3[(size_t)edst[e]*3 + w], cutb[e]*(APOT*acc));
}

__global__ void finalize_kernel(const float* __restrict__ att, const float* __restrict__ pot3,
                                float* __restrict__ feat, int Ntot){
  int gid = blockIdx.x*blockDim.x + threadIdx.x;
  if (gid >= Ntot*60) return;
  int n = gid/60, c = gid%60;
  float f = att[gid];
  if (c < 3){
    float p0 = pot3[(size_t)n*3+0], p1 = pot3[(size_t)n*3+1], p2 = pot3[(size_t)n*3+2];
    float curl = (c==0) ? (p2-p1) : (c==1) ? (p0-p2) : (p1-p0);
    f += 0.1f*curl;
  }
  feat[gid] = f;
}

__global__ void out_kernel(const float* __restrict__ feat, const float* __restrict__ Wout,
                           const float* __restrict__ wf, const float* __restrict__ bf,
                           float* __restrict__ out, int Ntot){
  int n = blockIdx.x*blockDim.x + threadIdx.x;
  if (n >= Ntot) return;
  float s = 0.f;
  for (int c=0;c<16;++c) s += feat[(size_t)n*60+c]*Wout[c];
  out[n] = (s*0.25f)*wf[0] + bf[0];
}

// ---------------- host launcher ----------------
static inline int cdiv(int a, int b){ return (a+b-1)/b; }

extern "C" void kernel_launch(void* const* d_in, const int* in_sizes, int n_in,
                              void* d_out, int out_size, void* d_ws, size_t ws_size,
                              hipStream_t stream) {
  const float* x    = (const float*)d_in[0];
  const float* pos  = (const float*)d_in[1];
  const int*   esrc = (const int*)  d_in[2];
  const int*   edst = (const int*)  d_in[3];
  const float* W_in = (const float*)d_in[4];
  const float* hq0  = (const float*)d_in[5];
  const float* hq1  = (const float*)d_in[6];
  const float* hq2  = (const float*)d_in[7];
  const float* fck1 = (const float*)d_in[8];
  const float* fck2 = (const float*)d_in[9];
  const float* fcv1 = (const float*)d_in[10];
  const float* fcv2 = (const float*)d_in[11];
  const float* dotw = (const float*)d_in[12];
  const float* fcp1 = (const float*)d_in[13];
  const float* fcp2 = (const float*)d_in[14];
  const float* Wout = (const float*)d_in[15];
  const float* w_f  = (const float*)d_in[16];
  const float* b_f  = (const float*)d_in[17];

  int E    = in_sizes[2];
  int Ntot = in_sizes[1]/3;
  int F    = in_sizes[0]/Ntot;

  float* p = (float*)d_ws;
  float* cg   = p; p += 384;
  float* feat = p; p += (size_t)Ntot*60;
  float* qb   = p; p += (size_t)Ntot*60;
  float* att  = p; p += (size_t)Ntot*60;   // att, pot3, z contiguous -> single zero fill
  float* pot3 = p; p += (size_t)Ntot*3;
  float* z    = p; p += (size_t)Ntot;
  float* eshb = p; p += (size_t)E*9;
  float* embb = p; p += (size_t)E*10;
  float* cutb = p; p += (size_t)E;
  float* hk   = p; p += (size_t)E*32;
  float* hv   = p; p += (size_t)E*32;
  float* hp   = p; p += (size_t)E*64;
  float* kb   = p; p += (size_t)E*60;
  float* vb   = p; p += (size_t)E*60;
  float* expv = p; p += (size_t)E;

  init_cg_kernel<<<1, 32, 0, stream>>>(cg);
  feat_init_kernel<<<cdiv(Ntot*60,256), 256, 0, stream>>>(x, W_in, feat, Ntot, F);
  edge_prep_kernel<<<cdiv(E,256), 256, 0, stream>>>(pos, esrc, edst, eshb, embb, cutb, E);

  for (int l=0; l<3; ++l){
    q_kernel<<<cdiv(Ntot*60,256), 256, 0, stream>>>(feat, hq0+l*256, hq1+l*64, hq2+l*16, qb, Ntot);
    h_kernel<<<cdiv(E*32,256), 256, 0, stream>>>(embb, fck1+l*320, hk, E, 32, 0.17677669529663687f);
    h_kernel<<<cdiv(E*32,256), 256, 0, stream>>>(embb, fcv1+l*320, hv, E, 32, 0.17677669529663687f);
    h_kernel<<<cdiv(E*64,256), 256, 0, stream>>>(embb, fcp1+l*640, hp, E, 64, 0.125f);
    fill_kernel<<<cdiv(Ntot*64,256), 256, 0, stream>>>(att, Ntot*64, 0.f);
    tp_kv_kernel<<<cdiv(E,16), 256, 0, stream>>>(feat, esrc, eshb, hk, fck2+l*27648, cg, kb, E);
    tp_kv_kernel<<<cdiv(E,16), 256, 0, stream>>>(feat, esrc, eshb, hv, fcv2+l*27648, cg, vb, E);
    logit_kernel<<<cdiv(E,128), 128, 0, stream>>>(qb, kb, edst, cutb, dotw+l*336, cg, expv, z, E);
    att_kernel<<<cdiv(E*60,256), 256, 0, stream>>>(vb, expv, z, edst, att, E);
    pot_kernel<<<cdiv(E*3,128), 128, 0, stream>>>(att, esrc, edst, eshb, hp, fcp2+l*94208, cutb, cg, pot3, E);
    finalize_kernel<<<cdiv(Ntot*60,256), 256, 0, stream>>>(att, pot3, feat, Ntot);
  }
  out_kernel<<<cdiv(Ntot,256), 256, 0, stream>>>(feat, Wout, w_f, b_f, (float*)d_out, Ntot);
}